// StackRNNCore_85315230368210
// MI455X (gfx1250) — compile-verified
//
#include <hip/hip_runtime.h>
#include <cstddef>

// ---------------------------------------------------------------------------
// StackRNNCore for MI455X (gfx1250, wave32).
// B=512, IN=256, H=512, NS=4, SS=128, CS=128, IC=768, K_total=1280.
// Bandwidth-bound op (~270 MB stack traffic @ 23.3 TB/s ~= 11.6 us);
// GEMMs done with V_WMMA_F32_16X16X4_F32 at full fp32 precision.
// Streaming stack blend uses non-temporal (TH=NT) b128 loads/stores so the
// once-touched 268 MB does not evict the reused weights/h from the 192MB L2.
// ---------------------------------------------------------------------------

#define B_   512
#define IN_  256
#define H_   512
#define NS_  4
#define SS_  128
#define CS_  128

typedef __attribute__((ext_vector_type(2))) float v2f;
typedef __attribute__((ext_vector_type(4))) float v4f;
typedef __attribute__((ext_vector_type(8))) float v8f;

// ---------------------------------------------------------------------------
// Kernel 1: h = tanh([inputs | top(stacks)] @ W_ih^T + core @ W_hh^T + biases)
// One wave per 16x16 output tile. K streamed in segments with f32 WMMA.
// A-frag per lane: row = m_tile*16 + lane%16, K = k0 + 2*(lane/16) + {0,1}
// B-frag per lane: col = n_tile*16 + lane%16, same K pair
// C/D: VGPR v holds row v + 8*(lane/16), col lane%16.
// ---------------------------------------------------------------------------
__global__ __launch_bounds__(128) void k_h(
    const float* __restrict__ inputs,      // (B, IN)
    const float* __restrict__ stacks,      // (B, NS, SS, CS)
    const float* __restrict__ core_state,  // (B, H)
    const float* __restrict__ W_ih,        // (H, IC=768) row-major
    const float* __restrict__ b_ih,        // (H)
    const float* __restrict__ W_hh,        // (H, H)
    const float* __restrict__ b_hh,        // (H)
    float* __restrict__ h_out)             // (B, H)
{
    const int lane   = threadIdx.x & 31;
    const int wave   = threadIdx.x >> 5;
    const int n_tile = blockIdx.x * 4 + wave;   // H/16 = 32 tiles, 4 waves/block
    const int m_tile = blockIdx.y;              // B/16 = 32 tiles
    const int g      = lane >> 4;               // half-wave group
    const int mn     = lane & 15;
    const int row    = m_tile * 16 + mn;        // A-matrix row this lane loads
    const int col    = n_tile * 16 + mn;        // B-matrix column this lane loads

    v8f acc = {};

    // --- segment 1: inputs, K in [0, 256) -----------------------------------
    {
        const float* ap = inputs + (size_t)row * IN_ + 2 * g;
        const float* bp = W_ih   + (size_t)col * 768 + 2 * g;
        #pragma unroll 4
        for (int k0 = 0; k0 < IN_; k0 += 4) {
            v2f a = *(const v2f*)(ap + k0);
            v2f b = *(const v2f*)(bp + k0);
            acc = __builtin_amdgcn_wmma_f32_16x16x4_f32(
                false, a, false, b, (short)0, acc, false, false);
        }
    }
    // --- segment 2: top-of-stacks, K in [256, 768) ---------------------------
    // top[b, ns*CS + c] = stacks[b, ns, 0, c]; split per ns so each sub-loop
    // has a constant base (pure constant-offset b64 loads, no VALU addr math).
    #pragma unroll
    for (int ns = 0; ns < NS_; ++ns) {
        const float* ap = stacks + (size_t)row * (NS_ * SS_ * CS_)
                                 + (size_t)ns * (SS_ * CS_) + 2 * g;
        const float* bp = W_ih + (size_t)col * 768 + 256 + ns * CS_ + 2 * g;
        #pragma unroll 4
        for (int k0 = 0; k0 < CS_; k0 += 4) {
            v2f a = *(const v2f*)(ap + k0);
            v2f b = *(const v2f*)(bp + k0);
            acc = __builtin_amdgcn_wmma_f32_16x16x4_f32(
                false, a, false, b, (short)0, acc, false, false);
        }
    }
    // --- segment 3: core_state, K in [768, 1280) -----------------------------
    {
        const float* ap = core_state + (size_t)row * H_ + 2 * g;
        const float* bp = W_hh       + (size_t)col * H_ + 2 * g;
        #pragma unroll 4
        for (int k0 = 0; k0 < H_; k0 += 4) {
            v2f a = *(const v2f*)(ap + k0);
            v2f b = *(const v2f*)(bp + k0);
            acc = __builtin_amdgcn_wmma_f32_16x16x4_f32(
                false, a, false, b, (short)0, acc, false, false);
        }
    }

    // --- epilogue: bias + tanh ----------------------------------------------
    const float bias = b_ih[col] + b_hh[col];
    #pragma unroll
    for (int v = 0; v < 8; ++v) {
        int r = m_tile * 16 + v + 8 * g;
        h_out[(size_t)r * H_ + col] = tanhf(acc[v] + bias);
    }
}

// ---------------------------------------------------------------------------
// Kernel 2: push = h @ W_push^T + b_push   (512 x 512 x 512)
// ---------------------------------------------------------------------------
__global__ __launch_bounds__(128) void k_push(
    const float* __restrict__ h,        // (B, H)
    const float* __restrict__ W_push,   // (NS*CS=512, H) row-major
    const float* __restrict__ b_push,   // (NS*CS)
    float* __restrict__ push_out)       // (B, NS*CS)
{
    const int lane   = threadIdx.x & 31;
    const int wave   = threadIdx.x >> 5;
    const int n_tile = blockIdx.x * 4 + wave;
    const int m_tile = blockIdx.y;
    const int g      = lane >> 4;
    const int mn     = lane & 15;
    const int row    = m_tile * 16 + mn;
    const int col    = n_tile * 16 + mn;

    v8f acc = {};
    const float* ap = h      + (size_t)row * H_ + 2 * g;
    const float* bp = W_push + (size_t)col * H_ + 2 * g;
    #pragma unroll 4
    for (int k0 = 0; k0 < H_; k0 += 4) {
        v2f a = *(const v2f*)(ap + k0);
        v2f b = *(const v2f*)(bp + k0);
        acc = __builtin_amdgcn_wmma_f32_16x16x4_f32(
            false, a, false, b, (short)0, acc, false, false);
    }
    const float bias = b_push[col];
    #pragma unroll
    for (int v = 0; v < 8; ++v) {
        int r = m_tile * 16 + v + 8 * g;
        push_out[(size_t)r * (NS_ * CS_) + col] = acc[v] + bias;
    }
}

// ---------------------------------------------------------------------------
// Kernel 3: act = softmax(h @ W_act^T + b_act) per (b, stack) triple.
// One wave32 per (b, s); lane-split dot product + shfl_xor reduction.
// ---------------------------------------------------------------------------
__global__ __launch_bounds__(256) void k_act(
    const float* __restrict__ h,       // (B, H)
    const float* __restrict__ W_act,   // (NS*3, H)
    const float* __restrict__ b_act,   // (NS*3)
    float* __restrict__ act_out)       // (B*NS, 3) = (pa, po, no)
{
    const int lane = threadIdx.x & 31;
    const int wave = threadIdx.x >> 5;
    const int task = blockIdx.x * 8 + wave;   // 0 .. B*NS-1
    const int b    = task >> 2;
    const int s    = task & 3;

    const float* hr = h + (size_t)b * H_;
    const float* w0 = W_act + (size_t)(s * 3 + 0) * H_;
    const float* w1 = W_act + (size_t)(s * 3 + 1) * H_;
    const float* w2 = W_act + (size_t)(s * 3 + 2) * H_;

    float s0 = 0.f, s1 = 0.f, s2 = 0.f;
    for (int k = lane; k < H_; k += 32) {
        float hv = hr[k];
        s0 = fmaf(hv, w0[k], s0);
        s1 = fmaf(hv, w1[k], s1);
        s2 = fmaf(hv, w2[k], s2);
    }
    #pragma unroll
    for (int off = 16; off > 0; off >>= 1) {
        s0 += __shfl_xor(s0, off, 32);
        s1 += __shfl_xor(s1, off, 32);
        s2 += __shfl_xor(s2, off, 32);
    }
    if (lane == 0) {
        float l0 = s0 + b_act[s * 3 + 0];
        float l1 = s1 + b_act[s * 3 + 1];
        float l2 = s2 + b_act[s * 3 + 2];
        float m  = fmaxf(l0, fmaxf(l1, l2));
        float e0 = expf(l0 - m), e1 = expf(l1 - m), e2 = expf(l2 - m);
        float inv = 1.f / (e0 + e1 + e2);
        act_out[task * 3 + 0] = e0 * inv;
        act_out[task * 3 + 1] = e1 * inv;
        act_out[task * 3 + 2] = e2 * inv;
    }
}

// ---------------------------------------------------------------------------
// Kernel 4: stack blend (bandwidth-critical, ~270 MB streaming).
//   d==0 : new = pa*push        + po*old[1]   + no*old[0]
//   d>=1 : new = pa*old[d-1]    + po*old[d+1] + no*old[d]   (old[SS] := 0)
// One block per (b,s). Thread = (v4f column, depth-group of 16 rows),
// rolling prev/cur/next registers -> stacks read ~1.11x, written 1x.
// Non-temporal b128 loads/stores: once-touched stream must not evict the
// reused weights/h from L2 (ISA section 7.3 TH=NT policy).
// ---------------------------------------------------------------------------
__device__ __forceinline__ v4f blend3(float pa, v4f x, float po, v4f y,
                                      float no, v4f z) {
    return pa * x + po * y + no * z;
}

__global__ __launch_bounds__(256) void k_stack(
    const float* __restrict__ stacks,   // (B, NS, SS, CS)
    const float* __restrict__ push,     // (B, NS*CS)
    const float* __restrict__ act,      // (B*NS, 3)
    float* __restrict__ out_stacks)     // (B, NS, SS, CS)
{
    const int bs = blockIdx.x;          // b*NS + s
    const int b  = bs >> 2;
    const int s  = bs & 3;
    const int c4 = threadIdx.x & 31;    // v4f column, CS/4 = 32
    const int dg = threadIdx.x >> 5;    // depth group 0..7 (16 rows each)

    const float pa = act[bs * 3 + 0];
    const float po = act[bs * 3 + 1];
    const float no = act[bs * 3 + 2];

    const v4f* old4 = (const v4f*)(stacks     + (size_t)bs * (SS_ * CS_));
    v4f*       out4 = (v4f*)      (out_stacks + (size_t)bs * (SS_ * CS_));
    const int d0 = dg * 16;

    if (d0 == 0) {
        v4f vcur  = __builtin_nontemporal_load(old4 + c4);
        v4f vnext = __builtin_nontemporal_load(old4 + 32 + c4);
        v4f p4 = ((const v4f*)(push + (size_t)b * (NS_ * CS_) + s * CS_))[c4];
        __builtin_nontemporal_store(blend3(pa, p4, po, vnext, no, vcur),
                                    out4 + c4);
        v4f vprev = vcur;
        vcur = vnext;
        #pragma unroll 5
        for (int d = 1; d < 16; ++d) {
            v4f vn = __builtin_nontemporal_load(old4 + (d + 1) * 32 + c4);
            __builtin_nontemporal_store(blend3(pa, vprev, po, vn, no, vcur),
                                        out4 + d * 32 + c4);
            vprev = vcur;
            vcur  = vn;
        }
    } else {
        v4f vprev = __builtin_nontemporal_load(old4 + (d0 - 1) * 32 + c4);
        v4f vcur  = __builtin_nontemporal_load(old4 + d0 * 32 + c4);
        #pragma unroll 4
        for (int d = d0; d < d0 + 16; ++d) {
            v4f vn = (d + 1 < SS_)
                   ? __builtin_nontemporal_load(old4 + (d + 1) * 32 + c4)
                   : (v4f){};
            __builtin_nontemporal_store(blend3(pa, vprev, po, vn, no, vcur),
                                        out4 + d * 32 + c4);
            vprev = vcur;
            vcur  = vn;
        }
    }
}

// ---------------------------------------------------------------------------
// Launcher
// ---------------------------------------------------------------------------
extern "C" void kernel_launch(void* const* d_in, const int* in_sizes, int n_in,
                              void* d_out, int out_size, void* d_ws, size_t ws_size,
                              hipStream_t stream) {
    const float* inputs     = (const float*)d_in[0];
    const float* stacks     = (const float*)d_in[1];
    const float* core_state = (const float*)d_in[2];
    const float* W_ih       = (const float*)d_in[3];
    const float* b_ih       = (const float*)d_in[4];
    const float* W_hh       = (const float*)d_in[5];
    const float* b_hh       = (const float*)d_in[6];
    const float* W_push     = (const float*)d_in[7];
    const float* b_push     = (const float*)d_in[8];
    const float* W_act      = (const float*)d_in[9];
    const float* b_act      = (const float*)d_in[10];

    float* out_h      = (float*)d_out;                 // (B, H)
    float* out_stacks = out_h + (size_t)B_ * H_;       // (B, NS, SS, CS)

    float* ws_push = (float*)d_ws;                     // (B, NS*CS)  1 MB
    float* ws_act  = ws_push + (size_t)B_ * NS_ * CS_; // (B*NS, 3)   24 KB

    // h = tanh(x W_ih^T + core W_hh^T + biases)
    k_h<<<dim3(H_ / 64, B_ / 16), 128, 0, stream>>>(
        inputs, stacks, core_state, W_ih, b_ih, W_hh, b_hh, out_h);

    // push = h W_push^T + b_push
    k_push<<<dim3((NS_ * CS_) / 64, B_ / 16), 128, 0, stream>>>(
        out_h, W_push, b_push, ws_push);

    // act = softmax(h W_act^T + b_act)
    k_act<<<(B_ * NS_) / 8, 256, 0, stream>>>(out_h, W_act, b_act, ws_act);

    // stack blend (bandwidth roofline: ~270 MB @ 23.3 TB/s)
    k_stack<<<B_ * NS_, 256, 0, stream>>>(stacks, ws_push, ws_act, out_stacks);
}